// Model_23484881174856
// MI455X (gfx1250) — compile-verified
//
#include <hip/hip_runtime.h>
#include <hip/hip_bf16.h>
#include <math.h>

// Problem constants (from reference)
#define NB   64      // batch
#define NA   32      // keyphrase groups per batch
#define KP   50      // keyphrases per group (MAX_KP)
#define DIM  128     // embedding dim
#define EPS  1e-8f

typedef __attribute__((ext_vector_type(2))) float v2f;
typedef __attribute__((ext_vector_type(8))) float v8f;

// -------------------------------------------------------------------------
// Kernel 1: gather + mean-pool + L2-normalize.
// One wave (32 lanes) per output row; lane owns 4 contiguous dims (float4).
// Output: normalized embeddings, [3][NB*NA][DIM] f32 in workspace.
// -------------------------------------------------------------------------
__global__ __launch_bounds__(256) void embed_pool_norm(
    const int* __restrict__ src_idx,
    const int* __restrict__ pos_idx,
    const int* __restrict__ neg_idx,
    const float* __restrict__ table,   // [VOCAB+1][DIM]
    float* __restrict__ outN)          // [3][NB*NA][DIM]
{
    const int ROWS = NB * NA;                              // 2048
    int wave = (blockIdx.x * blockDim.x + threadIdx.x) >> 5;
    int lane = threadIdx.x & 31;
    if (wave >= 3 * ROWS) return;

    int t = wave / ROWS;                                   // 0=src,1=pos,2=neg
    int r = wave - t * ROWS;
    const int* __restrict__ idx =
        (t == 0) ? src_idx : (t == 1) ? pos_idx : neg_idx;
    idx += (size_t)r * KP;

    float ax = 0.f, ay = 0.f, az = 0.f, aw = 0.f;
    #pragma unroll 5
    for (int k = 0; k < KP; ++k) {
        int id = idx[k];
        const float4 e =
            *(const float4*)(table + (size_t)id * DIM + lane * 4);
        ax += e.x; ay += e.y; az += e.z; aw += e.w;
    }
    const float inv_kp = 1.0f / (float)KP;
    ax *= inv_kp; ay *= inv_kp; az *= inv_kp; aw *= inv_kp;

    // wave32 allreduce of sum of squares
    float ss = ax * ax + ay * ay + az * az + aw * aw;
    #pragma unroll
    for (int off = 16; off >= 1; off >>= 1)
        ss += __shfl_xor(ss, off, 32);

    float n   = sqrtf(ss);
    float inv = 1.0f / fmaxf(n, EPS);                      // x / max(||x||, eps)

    float4 o;
    o.x = ax * inv; o.y = ay * inv; o.z = az * inv; o.w = aw * inv;
    *(float4*)(outN + ((size_t)t * ROWS + (size_t)r) * DIM + lane * 4) = o;
}

// -------------------------------------------------------------------------
// Kernel 2: per-batch max cosine via fp32 WMMA (V_WMMA_F32_16X16X4_F32).
// Block = 8 waves: waves 0-3 -> four 16x16 tiles of src·pos^T,
//                  waves 4-7 -> four 16x16 tiles of src·neg^T.
// f32 A 16x4 layout: lane L holds A[L%16][2*(L>>4)+v], v in {0,1}.
// f32 B 4x16 layout mirrors with N=L%16 -> identical indexing into the
// row-major [32][128] matrix (B is the transposed operand, which is exactly
// the tgt matrix for src·tgt^T).
// -------------------------------------------------------------------------
__global__ __launch_bounds__(256) void wmma_maxcos(
    const float* __restrict__ embN,    // [3][NB*NA][DIM]
    float* __restrict__ partial)       // [NB] per-batch softplus loss
{
    __shared__ float tmax[8];

    int b    = blockIdx.x;
    int wave = threadIdx.x >> 5;
    int lane = threadIdx.x & 31;
    int half = wave >> 2;              // 0 = pos, 1 = neg
    int tile = wave & 3;
    int mr   = (tile >> 1) * 16;
    int nc   = (tile & 1) * 16;

    const size_t mat = (size_t)NA * DIM;                   // 4096 floats
    const float* __restrict__ srcN = embN + (size_t)b * mat;
    const float* __restrict__ tgtN =
        embN + (size_t)(1 + half) * NB * mat + (size_t)b * mat;

    int row = lane & 15;
    int kp2 = (lane >> 4) << 1;        // 0 or 2
    const float* ap = srcN + (size_t)(mr + row) * DIM + kp2;
    const float* bp = tgtN + (size_t)(nc + row) * DIM + kp2;

    v8f c = {0.f, 0.f, 0.f, 0.f, 0.f, 0.f, 0.f, 0.f};
    #pragma unroll 4
    for (int k0 = 0; k0 < DIM; k0 += 4) {
        v2f av = *(const v2f*)(ap + k0);
        v2f bv = *(const v2f*)(bp + k0);
        c = __builtin_amdgcn_wmma_f32_16x16x4_f32(
                /*neg_a=*/false, av, /*neg_b=*/false, bv,
                /*c_mod=*/(short)0, c, /*reuse_a=*/false, /*reuse_b=*/false);
    }

    // lane-local max over 8 accumulators (rows M = {0..7}+8*(lane>>4))
    float m = c[0];
    #pragma unroll
    for (int i = 1; i < 8; ++i) m = fmaxf(m, c[i]);
    // wave32 max-reduce
    #pragma unroll
    for (int off = 16; off >= 1; off >>= 1)
        m = fmaxf(m, __shfl_xor(m, off, 32));

    if (lane == 0) tmax[wave] = m;
    __syncthreads();

    if (threadIdx.x == 0) {
        float p = fmaxf(fmaxf(tmax[0], tmax[1]), fmaxf(tmax[2], tmax[3]));
        float n = fmaxf(fmaxf(tmax[4], tmax[5]), fmaxf(tmax[6], tmax[7]));
        float z = n - p;               // softplus(-(p-n)) = softplus(n-p)
        // numerically stable softplus(z) = max(z,0) + log1p(exp(-|z|))
        partial[b] = fmaxf(z, 0.f) + log1pf(expf(-fabsf(z)));
    }
}

// -------------------------------------------------------------------------
// Kernel 3: mean over batches -> scalar loss.
// -------------------------------------------------------------------------
__global__ __launch_bounds__(64) void finalize(
    const float* __restrict__ partial, float* __restrict__ out)
{
    __shared__ float s[NB];
    s[threadIdx.x] = partial[threadIdx.x];
    __syncthreads();
    if (threadIdx.x == 0) {
        float sum = 0.f;
        #pragma unroll
        for (int i = 0; i < NB; ++i) sum += s[i];
        out[0] = sum / (float)NB;
    }
}

// -------------------------------------------------------------------------
extern "C" void kernel_launch(void* const* d_in, const int* in_sizes, int n_in,
                              void* d_out, int out_size, void* d_ws, size_t ws_size,
                              hipStream_t stream)
{
    (void)in_sizes; (void)n_in; (void)out_size; (void)ws_size;

    const int*   src_idx = (const int*)d_in[0];   // [64,32,50]
    const int*   pos_idx = (const int*)d_in[1];   // [64,32,50]
    const int*   neg_idx = (const int*)d_in[2];   // [64,32,50]
    const float* table   = (const float*)d_in[3]; // [100001,128]
    float*       out     = (float*)d_out;         // scalar

    float* embN    = (float*)d_ws;                          // 3*2048*128 f32
    float* partial = embN + (size_t)3 * NB * NA * DIM;      // 64 f32

    // K1: 3*2048 waves, 8 waves/block -> 768 blocks
    const int total_waves = 3 * NB * NA;
    embed_pool_norm<<<(total_waves * 32 + 255) / 256, 256, 0, stream>>>(
        src_idx, pos_idx, neg_idx, table, embN);

    // K2: one block per batch
    wmma_maxcos<<<NB, 256, 0, stream>>>(embN, partial);

    // K3: scalar reduction
    finalize<<<1, 64, 0, stream>>>(partial, out);
}